// SpatialGeneralConv_36421322670216
// MI455X (gfx1250) — compile-verified
//
#include <hip/hip_runtime.h>
#include <hip/hip_bf16.h>

// HGT layer for MI455X (gfx1250, wave32, WMMA).
// f32 end-to-end: pipeline is memory/atomic bound, so keep full precision and
// put ALL matrix math (per-type projections AND per-relation edge transforms)
// on V_WMMA_F32_16X16X4_F32.

typedef float v2f __attribute__((ext_vector_type(2)));
typedef float v8f __attribute__((ext_vector_type(8)));

#define D128 128
#define HEADS 8
#define DK 16
#define NTYPES 4
#define NREL 4

// ---------------------------------------------------------------- init ------
__global__ __launch_bounds__(256) void init_kernel(int* __restrict__ segmax,
                                                   float* __restrict__ denom,
                                                   float* __restrict__ aggr,
                                                   int nseg, int naggr) {
  int i = blockIdx.x * 256 + threadIdx.x;
  int stride = gridDim.x * 256;
  for (int idx = i; idx < nseg; idx += stride) {
    segmax[idx] = (int)0x80000000;  // ordered-int key for -inf
    denom[idx] = 0.0f;
  }
  for (int idx = i; idx < naggr; idx += stride) aggr[idx] = 0.0f;
}

// ------------------------------------------------- per-type projections -----
// grid: (ceil(N/16), 3), block: 256 (8 waves; wave w -> columns [16w,16w+16)).
// For each of 4 node types: C_t = X_tile @ W[t] via 32 chained
// V_WMMA_F32_16X16X4_F32, then per-row select by node type (rows live in the
// C-layout VGPR index + lane half -> pure cndmask select).
__global__ __launch_bounds__(256) void proj_kernel(
    const float* __restrict__ x, const int* __restrict__ node_type,
    const float* __restrict__ Wk, const float* __restrict__ bk,
    const float* __restrict__ Wq, const float* __restrict__ bq,
    const float* __restrict__ Wv, const float* __restrict__ bv,
    float* __restrict__ Ko, float* __restrict__ Qo, float* __restrict__ Vo,
    int n) {
  __shared__ float xs[16][132];  // pad 132: conflict-free A-frag reads
  __shared__ int ts[16];

  const float* W;
  const float* B;
  float* O;
  if (blockIdx.y == 0) { W = Wk; B = bk; O = Ko; }
  else if (blockIdx.y == 1) { W = Wq; B = bq; O = Qo; }
  else { W = Wv; B = bv; O = Vo; }

  int row0 = blockIdx.x * 16;
  int tid = threadIdx.x;
  for (int i = tid; i < 16 * D128; i += 256) {
    int r = i >> 7, c = i & 127;
    int gr = row0 + r;
    xs[r][c] = (gr < n) ? x[(size_t)gr * D128 + c] : 0.0f;
  }
  if (tid < 16) ts[tid] = (row0 + tid < n) ? node_type[row0 + tid] : 0;
  __syncthreads();

  int wave = tid >> 5;
  int lane = tid & 31;
  int half = lane >> 4;
  int l16 = lane & 15;
  int col0 = wave * 16;

  v8f res = {};
  for (int t = 0; t < NTYPES; ++t) {
    v8f c = {};
    const float* Wt = W + (size_t)t * D128 * D128;
    for (int kk = 0; kk < D128; kk += 4) {
      int ka = kk + half * 2;  // lane-half selects K pair; B mirrors A
      v2f a, b;
      a.x = xs[l16][ka];
      a.y = xs[l16][ka + 1];
      b.x = Wt[(size_t)ka * D128 + col0 + l16];
      b.y = Wt[(size_t)(ka + 1) * D128 + col0 + l16];
      c = __builtin_amdgcn_wmma_f32_16x16x4_f32(false, a, false, b,
                                                (short)0, c, false, false);
    }
    float bt = B[t * D128 + col0 + l16];
#pragma unroll
    for (int i = 0; i < 8; ++i) {
      int row = i + 8 * half;
      if (ts[row] == t) res[i] = c[i] + bt;
    }
  }
#pragma unroll
  for (int i = 0; i < 8; ++i) {
    int row = row0 + i + 8 * half;
    if (row < n) O[(size_t)row * D128 + col0 + l16] = res[i];
  }
}

// -------------------------------------------------- edge logits + segmax ----
// Wave per 16-edge tile. k_rel = K_tile(16x16 per head) @ rel_att[r][h] on the
// WMMA pipe for all 4 relations, per-row select by edge relation, then
// row-wise dot with gathered Q[dst] + 16-lane shfl reduce, ordered-int
// atomic max into seg_max. block = 128 (4 waves), LDS ~34KB.
__global__ __launch_bounds__(128) void edge_logits_kernel(
    const int* __restrict__ ei, const int* __restrict__ et,
    const float* __restrict__ Kf, const float* __restrict__ Qf,
    const float* __restrict__ rel_att, const float* __restrict__ rel_pri,
    float* __restrict__ logits, int* __restrict__ segmax, int E_) {
  __shared__ float kb[4][16][132];
  __shared__ int dst_s[4][16];
  __shared__ int rel_s[4][16];

  int w = threadIdx.x >> 5;
  int lane = threadIdx.x & 31;
  int e0 = (blockIdx.x * 4 + w) * 16;

  if (lane < 16) {
    int e = e0 + lane;
    bool act = (e < E_);
    int ec = act ? e : (E_ - 1);
    dst_s[w][lane] = act ? ei[E_ + ec] : -1;  // -1 marks padding row
    rel_s[w][lane] = et[ec];
  }
  for (int i = 0; i < 16; ++i) {
    int e = e0 + i;
    int ec = (e < E_) ? e : (E_ - 1);
    int src = ei[ec];  // wave-uniform broadcast load
    *(float4*)(&kb[w][i][lane * 4]) =
        *(const float4*)(Kf + (size_t)src * D128 + lane * 4);
  }
  __syncthreads();

  int half = lane >> 4, l16 = lane & 15;
  for (int h = 0; h < HEADS; ++h) {
    v8f sel = {};
    for (int r = 0; r < NREL; ++r) {
      v8f c = {};
      const float* A = rel_att + ((size_t)r * HEADS + h) * (DK * DK);
#pragma unroll
      for (int kk = 0; kk < DK; kk += 4) {
        int ka = kk + half * 2;
        v2f a, b;
        a.x = kb[w][l16][h * DK + ka];
        a.y = kb[w][l16][h * DK + ka + 1];
        b.x = A[ka * DK + l16];
        b.y = A[(ka + 1) * DK + l16];
        c = __builtin_amdgcn_wmma_f32_16x16x4_f32(false, a, false, b,
                                                  (short)0, c, false, false);
      }
#pragma unroll
      for (int i = 0; i < 8; ++i) {
        int row = i + 8 * half;
        if (rel_s[w][row] == r) sel[i] = c[i];
      }
    }
    // logits: row-wise dot(k_rel, q) -> 16-lane reduce -> store + atomic max
#pragma unroll
    for (int i = 0; i < 8; ++i) {
      int row = i + 8 * half;
      int d = dst_s[w][row];
      int dc = (d >= 0) ? d : 0;
      float part = sel[i] * Qf[(size_t)dc * D128 + h * DK + l16];
      part += __shfl_xor(part, 1, 16);
      part += __shfl_xor(part, 2, 16);
      part += __shfl_xor(part, 4, 16);
      part += __shfl_xor(part, 8, 16);
      if (l16 == 0 && d >= 0) {
        int r = rel_s[w][row];
        float lg = part * rel_pri[r * HEADS + h] * 0.25f;  // 1/sqrt(DK)
        logits[(size_t)(e0 + row) * HEADS + h] = lg;
        int bi = __float_as_int(lg);
        int key = (bi >= 0) ? bi : (bi ^ 0x7fffffff);  // order-preserving
        atomicMax(&segmax[d * HEADS + h], key);
      }
    }
  }
}

// ----------------------------------------------------- exp + denominator ----
__global__ __launch_bounds__(256) void edge_exp_kernel(
    const int* __restrict__ ei, const int* __restrict__ segmax,
    float* __restrict__ logits /* in: logits, out: ex (in-place) */,
    float* __restrict__ denom, int E_) {
  int idx = blockIdx.x * 256 + threadIdx.x;
  if (idx >= E_ * HEADS) return;
  int e = idx >> 3;
  int h = idx & 7;
  int dst = ei[E_ + e];
  int key = segmax[dst * HEADS + h];
  float m = (key >= 0) ? __int_as_float(key) : __int_as_float(key ^ 0x7fffffff);
  float ex = expf(logits[idx] - m);
  logits[idx] = ex;
  atomicAdd(&denom[dst * HEADS + h], ex);
}

// ------------------------------------------------ message + aggregation -----
// Same WMMA-per-16-edge-tile template: msg = V_tile @ rel_msg[r][h] with
// relation select, scaled by pre-staged att, atomic-add into aggr.
__global__ __launch_bounds__(128) void edge_aggr_kernel(
    const int* __restrict__ ei, const int* __restrict__ et,
    const float* __restrict__ Vf, const float* __restrict__ rel_msg,
    const float* __restrict__ exbuf, const float* __restrict__ denom,
    float* __restrict__ aggr, int E_) {
  __shared__ float vb[4][16][132];
  __shared__ float att_s[4][16][HEADS];
  __shared__ int dst_s[4][16];
  __shared__ int rel_s[4][16];

  int w = threadIdx.x >> 5;
  int lane = threadIdx.x & 31;
  int e0 = (blockIdx.x * 4 + w) * 16;

  if (lane < 16) {
    int e = e0 + lane;
    bool act = (e < E_);
    int ec = act ? e : (E_ - 1);
    dst_s[w][lane] = act ? ei[E_ + ec] : -1;
    rel_s[w][lane] = et[ec];
  }
  for (int p = lane; p < 16 * HEADS; p += 32) {
    int row = p >> 3, h = p & 7;
    int e = e0 + row;
    bool act = (e < E_);
    int ec = act ? e : (E_ - 1);
    float at = 0.0f;
    if (act) {
      int d = ei[E_ + ec];
      at = exbuf[(size_t)ec * HEADS + h] / denom[d * HEADS + h];
    }
    att_s[w][row][h] = at;
  }
  for (int i = 0; i < 16; ++i) {
    int e = e0 + i;
    int ec = (e < E_) ? e : (E_ - 1);
    int src = ei[ec];
    *(float4*)(&vb[w][i][lane * 4]) =
        *(const float4*)(Vf + (size_t)src * D128 + lane * 4);
  }
  __syncthreads();

  int half = lane >> 4, l16 = lane & 15;
  for (int h = 0; h < HEADS; ++h) {
    v8f sel = {};
    for (int r = 0; r < NREL; ++r) {
      v8f c = {};
      const float* M = rel_msg + ((size_t)r * HEADS + h) * (DK * DK);
#pragma unroll
      for (int kk = 0; kk < DK; kk += 4) {
        int ka = kk + half * 2;
        v2f a, b;
        a.x = vb[w][l16][h * DK + ka];
        a.y = vb[w][l16][h * DK + ka + 1];
        b.x = M[ka * DK + l16];
        b.y = M[(ka + 1) * DK + l16];
        c = __builtin_amdgcn_wmma_f32_16x16x4_f32(false, a, false, b,
                                                  (short)0, c, false, false);
      }
#pragma unroll
      for (int i = 0; i < 8; ++i) {
        int row = i + 8 * half;
        if (rel_s[w][row] == r) sel[i] = c[i];
      }
    }
#pragma unroll
    for (int i = 0; i < 8; ++i) {
      int row = i + 8 * half;
      int d = dst_s[w][row];
      if (d >= 0) {
        atomicAdd(&aggr[(size_t)d * D128 + h * DK + l16],
                  sel[i] * att_s[w][row][h]);
      }
    }
  }
}

// ---------------------- GELU + Wa projection (WMMA) + skip + LayerNorm ------
__global__ __launch_bounds__(256) void post_kernel(
    const float* __restrict__ x, const int* __restrict__ node_type,
    const float* __restrict__ aggr, const float* __restrict__ Wa,
    const float* __restrict__ ba, const float* __restrict__ skip,
    const float* __restrict__ ln_g, const float* __restrict__ ln_b,
    float* __restrict__ out, int n) {
  __shared__ float hs[16][132];
  __shared__ float tr[16][132];
  __shared__ int ts[16];

  int row0 = blockIdx.x * 16;
  int tid = threadIdx.x;
  for (int i = tid; i < 16 * D128; i += 256) {
    int r = i >> 7, c = i & 127;
    int gr = row0 + r;
    float v = (gr < n) ? aggr[(size_t)gr * D128 + c] : 0.0f;
    hs[r][c] = 0.5f * v * (1.0f + erff(v * 0.70710678118654752f));  // exact gelu
  }
  if (tid < 16) ts[tid] = (row0 + tid < n) ? node_type[row0 + tid] : 0;
  __syncthreads();

  int wave = tid >> 5;
  int lane = tid & 31;
  int half = lane >> 4;
  int l16 = lane & 15;
  int col0 = wave * 16;

  v8f res = {};
  for (int t = 0; t < NTYPES; ++t) {
    v8f c = {};
    const float* Wt = Wa + (size_t)t * D128 * D128;
    for (int kk = 0; kk < D128; kk += 4) {
      int ka = kk + half * 2;
      v2f a, b;
      a.x = hs[l16][ka];
      a.y = hs[l16][ka + 1];
      b.x = Wt[(size_t)ka * D128 + col0 + l16];
      b.y = Wt[(size_t)(ka + 1) * D128 + col0 + l16];
      c = __builtin_amdgcn_wmma_f32_16x16x4_f32(false, a, false, b,
                                                (short)0, c, false, false);
    }
    float bt = ba[t * D128 + col0 + l16];
#pragma unroll
    for (int i = 0; i < 8; ++i) {
      int row = i + 8 * half;
      if (ts[row] == t) res[i] = c[i] + bt;
    }
  }
#pragma unroll
  for (int i = 0; i < 8; ++i) tr[i + 8 * half][col0 + l16] = res[i];
  __syncthreads();

  // LayerNorm: each aligned 16-lane group handles one node.
  int nodeL = tid >> 4;
  int cl = tid & 15;
  int grow = row0 + nodeL;
  int t = ts[nodeL];
  float alpha = 1.0f / (1.0f + expf(-skip[t]));
  float pre[8];
  float s = 0.0f, s2 = 0.0f;
#pragma unroll
  for (int j = 0; j < 8; ++j) {
    int c = cl + 16 * j;
    float xv = (grow < n) ? x[(size_t)grow * D128 + c] : 0.0f;
    float p = tr[nodeL][c] * alpha + xv * (1.0f - alpha);
    pre[j] = p;
    s += p;
    s2 += p * p;
  }
  s += __shfl_xor(s, 8, 16);  s2 += __shfl_xor(s2, 8, 16);
  s += __shfl_xor(s, 4, 16);  s2 += __shfl_xor(s2, 4, 16);
  s += __shfl_xor(s, 2, 16);  s2 += __shfl_xor(s2, 2, 16);
  s += __shfl_xor(s, 1, 16);  s2 += __shfl_xor(s2, 1, 16);
  float mu = s * (1.0f / 128.0f);
  float var = s2 * (1.0f / 128.0f) - mu * mu;
  float rstd = rsqrtf(var + 1e-5f);
  if (grow < n) {
#pragma unroll
    for (int j = 0; j < 8; ++j) {
      int c = cl + 16 * j;
      out[(size_t)grow * D128 + c] =
          (pre[j] - mu) * rstd * ln_g[t * D128 + c] + ln_b[t * D128 + c];
    }
  }
}

// ---------------------------------------------------------------- launch ----
extern "C" void kernel_launch(void* const* d_in, const int* in_sizes, int n_in,
                              void* d_out, int out_size, void* d_ws,
                              size_t ws_size, hipStream_t stream) {
  const float* x = (const float*)d_in[0];
  const int* node_type = (const int*)d_in[1];
  const int* edge_index = (const int*)d_in[2];
  const int* edge_type = (const int*)d_in[3];
  // d_in[4] edge_spatial: unused by the reference
  const float* Wk = (const float*)d_in[5];
  const float* bk = (const float*)d_in[6];
  const float* Wq = (const float*)d_in[7];
  const float* bq = (const float*)d_in[8];
  const float* Wv = (const float*)d_in[9];
  const float* bv = (const float*)d_in[10];
  const float* Wa = (const float*)d_in[11];
  const float* ba = (const float*)d_in[12];
  const float* rel_pri = (const float*)d_in[13];
  const float* rel_att = (const float*)d_in[14];
  const float* rel_msg = (const float*)d_in[15];
  const float* skip = (const float*)d_in[16];
  const float* ln_g = (const float*)d_in[17];
  const float* ln_b = (const float*)d_in[18];
  (void)n_in; (void)out_size; (void)ws_size;

  int N = in_sizes[1];
  int E = in_sizes[3];

  char* ws = (char*)d_ws;
  size_t off = 0;
  auto take = [&](size_t bytes) -> void* {
    void* p = ws + off;
    off += (bytes + 255) & ~(size_t)255;
    return p;
  };
  float* Kf = (float*)take((size_t)N * D128 * 4);
  float* Qf = (float*)take((size_t)N * D128 * 4);
  float* Vf = (float*)take((size_t)N * D128 * 4);
  float* logits = (float*)take((size_t)E * HEADS * 4);  // reused in-place as ex
  int* segmax = (int*)take((size_t)N * HEADS * 4);
  float* denom = (float*)take((size_t)N * HEADS * 4);
  float* aggr = (float*)take((size_t)N * D128 * 4);

  init_kernel<<<1024, 256, 0, stream>>>(segmax, denom, aggr, N * HEADS,
                                        N * D128);
  dim3 gp((N + 15) / 16, 3);
  proj_kernel<<<gp, 256, 0, stream>>>(x, node_type, Wk, bk, Wq, bq, Wv, bv, Kf,
                                      Qf, Vf, N);
  int tiles = (E + 15) / 16;
  int eblocks = (tiles + 3) / 4;
  edge_logits_kernel<<<eblocks, 128, 0, stream>>>(
      edge_index, edge_type, Kf, Qf, rel_att, rel_pri, logits, segmax, E);
  edge_exp_kernel<<<(E * HEADS + 255) / 256, 256, 0, stream>>>(
      edge_index, segmax, logits, denom, E);
  edge_aggr_kernel<<<eblocks, 128, 0, stream>>>(
      edge_index, edge_type, Vf, rel_msg, logits, denom, aggr, E);
  post_kernel<<<(N + 15) / 16, 256, 0, stream>>>(x, node_type, aggr, Wa, ba,
                                                 skip, ln_g, ln_b,
                                                 (float*)d_out, N);
}